// TopKGraph_19327352832063
// MI455X (gfx1250) — compile-verified
//
#include <hip/hip_runtime.h>
#include <hip/hip_bf16.h>
#include <stdint.h>

#define N_ELEM 16384
#define DDIM   256
#define KSEL   8192
#define JCHUNK 1024

typedef unsigned int v4u __attribute__((ext_vector_type(4)));
typedef int          v8i __attribute__((ext_vector_type(8)));
typedef int          v4i __attribute__((ext_vector_type(4)));

// Monotone bijection IEEE-754 f32 -> uint32 (ascending total order).
__device__ __forceinline__ uint32_t f32_ord(float f) {
  uint32_t b = __float_as_uint(f);
  return b ^ ((b >> 31) ? 0xFFFFFFFFu : 0x80000000u);
}
// 64-bit key: value-major (descending by >), index-minor (ascending: ~i).
// key_j > key_i  <=>  (s_j > s_i) || (s_j == s_i && j < i)  -- exact top_k order.
__device__ __forceinline__ uint64_t make_key(float f, int i) {
  return ((uint64_t)f32_ord(f) << 32) | (uint32_t)(~i);
}

// ---------------------------------------------------------------------------
// Phase 0: zero the rank counters.
// ---------------------------------------------------------------------------
__global__ __launch_bounds__(256) void zero_cnt_kernel(int* __restrict__ cnt_ws) {
  cnt_ws[blockIdx.x * 256 + threadIdx.x] = 0;
}

// ---------------------------------------------------------------------------
// Phase 1: partial ranks. Grid (64, 16): block (bi, bj) compares the 256
// elements of i-chunk bi against the 1024 keys of j-chunk bj (staged in LDS),
// accumulating into cnt_ws[i] with atomicAdd (order-independent int sum).
// One v_cmp_gt_u64 per candidate; ds_load_b128 broadcast fetches two keys.
// ---------------------------------------------------------------------------
__global__ __launch_bounds__(256) void rank_partial_kernel(
    const float* __restrict__ scores, int* __restrict__ cnt_ws) {
  __shared__ uint64_t jkeys[JCHUNK];                    // 8 KB
  const int tid = threadIdx.x;
  const int jbase = blockIdx.y * JCHUNK;
  for (int k = 0; k < JCHUNK / 256; ++k) {
    int j = jbase + tid + k * 256;
    jkeys[tid + k * 256] = make_key(scores[j], j);      // coalesced, L2-hot
  }
  __syncthreads();

  const int i = blockIdx.x * 256 + tid;
  const uint64_t ki = make_key(scores[i], i);
  int partial = 0;
#pragma unroll 8
  for (int jj = 0; jj < JCHUNK; ++jj)                   // uniform jj -> LDS broadcast
    partial += (int)(jkeys[jj] > ki);
  atomicAdd(&cnt_ws[i], partial);
}

// ---------------------------------------------------------------------------
// Phase 2: elements with rank < K land at output slot `rank` (sorted desc).
// Ranks are a bijection (keys all distinct), so each slot written once.
// ---------------------------------------------------------------------------
__global__ __launch_bounds__(256) void scatter_topk_kernel(
    const float* __restrict__ scores, const int* __restrict__ cnt_ws,
    int* __restrict__ idx_ws, float* __restrict__ val_ws,
    int* __restrict__ idx_out) {
  const int i = blockIdx.x * 256 + threadIdx.x;
  const int c = cnt_ws[i];
  if (c < KSEL) {
    idx_ws[c]  = i;
    val_ws[c]  = scores[i];
    idx_out[c] = i;                                     // idx output (int32 bits)
  }
}

// ---------------------------------------------------------------------------
// pooled_H[r, :] = H[idx[r], :] * values[r].  Fully coalesced.
// ---------------------------------------------------------------------------
__global__ __launch_bounds__(256) void pool_h_kernel(
    const float* __restrict__ H, const int* __restrict__ idx_ws,
    const float* __restrict__ val_ws, float* __restrict__ outH) {
  const int r = blockIdx.x;                             // 8192 rows
  const int t = threadIdx.x;                            // 256 cols
  const int src = idx_ws[r];
  const float v = val_ws[r];
  outH[(size_t)r * DDIM + t] = H[(size_t)src * DDIM + t] * v;
}

// ---------------------------------------------------------------------------
// pooled_A[b, j] = A[idx[b], idx[j]].  Stage the full 64 KB source row into
// LDS with one TDM tensor_load_to_lds, then gather columns from LDS and
// write coalesced float4 stores.  ~768 MB total HBM traffic => ~33 us floor.
// ---------------------------------------------------------------------------
__global__ __launch_bounds__(256) void pool_a_kernel(
    const float* __restrict__ A, const int* __restrict__ idx_ws,
    float* __restrict__ outA) {
  __shared__ float row[N_ELEM];                         // 64 KB (<= 320 KB/WGP)
  const int b = blockIdx.x;
  const int src_row = idx_ws[b];                        // uniform per block

  // --- Tensor DMA descriptor (D#), per CDNA5 ISA ---
  uint64_t ga = (uint64_t)(uintptr_t)(A + (size_t)src_row * N_ELEM);
  uint32_t lds_base = (uint32_t)(uintptr_t)(&row[0]);   // LDS offset (low 32b of flat)
  // Only wave 0 carries a valid descriptor; other waves get count=0 (NULL
  // tensor) so exactly one real 64 KB load issues even if the compiler
  // exec-masks instead of branching (TDM issues regardless of EXEC).
  uint32_t cnt = (threadIdx.x < 32) ? 1u : 0u;

  v4u g0;
  g0.x = cnt;                                           // count | is_restore=0 | gather=0
  g0.y = lds_base;                                      // lds_addr[31:0]
  g0.z = (uint32_t)ga;                                  // global_addr[31:0]
  g0.w = (uint32_t)((ga >> 32) & 0x1FFFFFFu)            // global_addr[56:32]
         | (2u << 30);                                  // type = 2 ("image")

  v8i g1;
  g1[0] = 0x00020000;                                   // data_size=2 (4B); wg_mask=0
  g1[1] = (int)(16384u << 16);                          // tensor_dim0[15:0] @ bits63:48
  g1[2] = (int)(1u << 16);                              // tensor_dim0 hi=0, tensor_dim1=1
  g1[3] = (int)(16384u << 16);                          // tile_dim0=16384 @ bits127:112
  g1[4] = 1;                                            // tile_dim1=1, tile_dim2=0
  g1[5] = 16384;                                        // tensor_dim0_stride[31:0]
  g1[6] = 0;
  g1[7] = 0;
  v4i g2 = {0, 0, 0, 0};
  v4i g3 = {0, 0, 0, 0};
  v8i g4 = {0, 0, 0, 0, 0, 0, 0, 0};                    // extra group (unused)

  if (threadIdx.x < 32) {
    __builtin_amdgcn_tensor_load_to_lds(g0, g1, g2, g3, g4, /*cpol=*/0);
  }
  __builtin_amdgcn_s_wait_tensorcnt(0);
  __syncthreads();

  float* orow = outA + (size_t)b * KSEL;
#pragma unroll
  for (int it = 0; it < KSEL / (256 * 4); ++it) {       // 8 iterations
    int e = it * 256 + threadIdx.x;
    int4 c = reinterpret_cast<const int4*>(idx_ws)[e];  // coalesced, L2-hot
    float4 r;
    r.x = row[c.x];
    r.y = row[c.y];
    r.z = row[c.z];
    r.w = row[c.w];
    reinterpret_cast<float4*>(orow)[e] = r;             // coalesced b128 store
  }
}

// ---------------------------------------------------------------------------
extern "C" void kernel_launch(void* const* d_in, const int* in_sizes, int n_in,
                              void* d_out, int out_size, void* d_ws, size_t ws_size,
                              hipStream_t stream) {
  const float* scores = (const float*)d_in[0];          // (16384,)
  const float* H      = (const float*)d_in[1];          // (16384, 256)
  const float* A      = (const float*)d_in[2];          // (16384, 16384)
  // d_in[3] = pooling_size (constant 8192)

  float* out    = (float*)d_out;
  float* outH   = out;                                  // 8192*256
  float* outA   = out + (size_t)KSEL * DDIM;            // 8192*8192
  int*   outIdx = (int*)(out + (size_t)KSEL * DDIM + (size_t)KSEL * KSEL);

  int*   idx_ws = (int*)d_ws;                                        // 32 KB
  float* val_ws = (float*)((char*)d_ws + (size_t)KSEL * 4);          // 32 KB
  int*   cnt_ws = (int*)((char*)d_ws + (size_t)KSEL * 8);            // 64 KB

  zero_cnt_kernel<<<N_ELEM / 256, 256, 0, stream>>>(cnt_ws);
  rank_partial_kernel<<<dim3(N_ELEM / 256, N_ELEM / JCHUNK), 256, 0, stream>>>(scores, cnt_ws);
  scatter_topk_kernel<<<N_ELEM / 256, 256, 0, stream>>>(scores, cnt_ws, idx_ws, val_ws, outIdx);
  pool_h_kernel<<<KSEL, 256, 0, stream>>>(H, idx_ws, val_ws, outH);
  pool_a_kernel<<<KSEL, 256, 0, stream>>>(A, idx_ws, outA);
}